// SGCNet_25598005084527
// MI455X (gfx1250) — compile-verified
//
#include <hip/hip_runtime.h>

typedef float v2f __attribute__((ext_vector_type(2)));
typedef float v8f __attribute__((ext_vector_type(8)));

// ---------------------------------------------------------------------------
// 1) degree init (self-loop contributes 1 to every node)
// ---------------------------------------------------------------------------
__global__ void sgc_deg_init(float* __restrict__ deg, int n) {
    int i = blockIdx.x * blockDim.x + threadIdx.x;
    if (i < n) deg[i] = 1.0f;
}

// 2) degree count over dst
__global__ void sgc_deg_count(const long long* __restrict__ dst,
                              float* __restrict__ deg, int e) {
    int i = blockIdx.x * blockDim.x + threadIdx.x;
    if (i < e) unsafeAtomicAdd(&deg[(int)dst[i]], 1.0f);
}

// 3) dinv = deg > 0 ? rsqrt(deg) : 0   (in place)
__global__ void sgc_dinv(float* __restrict__ deg, int n) {
    int i = blockIdx.x * blockDim.x + threadIdx.x;
    if (i < n) {
        float d = deg[i];
        deg[i] = (d > 0.0f) ? __frsqrt_rn(d) : 0.0f;
    }
}

// 4) per-edge symmetric norm, computed once, reused for all K hops
__global__ void sgc_norm(const long long* __restrict__ src,
                         const long long* __restrict__ dst,
                         const float* __restrict__ dinv,
                         float* __restrict__ norm, int e) {
    int i = blockIdx.x * blockDim.x + threadIdx.x;
    if (i < e) norm[i] = dinv[(int)src[i]] * dinv[(int)dst[i]];
}

// ---------------------------------------------------------------------------
// 5) z = x @ w  via V_WMMA_F32_16X16X4_F32 (exact fp32 matrix path).
//    One wave handles a 16-row tile; K=128 consumed in 32 chunks of 4.
//    A (16x4): lanes 0-15 = rows, VGPR0/1 = K0/K1; lanes 16-31 = K2/K3.
//    B (4x16): w replicated into all 16 columns -> every D column = x_tile@w.
//    D (16x16): VGPR r holds M = r (lanes 0-15) / r+8 (lanes 16-31).
// ---------------------------------------------------------------------------
__global__ __launch_bounds__(256) void sgc_gemv_wmma(const float* __restrict__ x,
                                                     const float* __restrict__ w,
                                                     float* __restrict__ z, int n) {
    const int wave = (int)((blockIdx.x * blockDim.x + threadIdx.x) >> 5);
    const int lane = threadIdx.x & 31;
    const int row0 = wave * 16;
    if (row0 >= n) return;                 // wave-uniform guard (EXEC stays full)

    const int m  = lane & 15;              // row within tile
    const int kh = lane >> 4;              // K half-pair selector (0: K0/1, 1: K2/3)
    const float* __restrict__ xr = x + (size_t)(row0 + m) * 128 + kh * 2;
    const float* __restrict__ wr = w + kh * 2;

    v8f acc = {};
#pragma unroll
    for (int kb = 0; kb < 128; kb += 4) {
        v2f a, b;
        a.x = xr[kb];     a.y = xr[kb + 1];
        b.x = wr[kb];     b.y = wr[kb + 1];
        acc = __builtin_amdgcn_wmma_f32_16x16x4_f32(
            /*neg_a=*/false, a, /*neg_b=*/false, b,
            /*c_mod=*/(short)0, acc, /*reuse_a=*/false, /*reuse_b=*/false);
    }

    if (m == 0) {                          // lane 0 -> rows 0-7, lane 16 -> rows 8-15
        const int base = row0 + kh * 8;
#pragma unroll
        for (int r = 0; r < 8; ++r) z[base + r] = acc[r];
    }
}

// ---------------------------------------------------------------------------
// 6) one propagation hop on the projected scalar:
//    zn[i] = dinv[i]^2 * zc[i]  (self loop), then scatter-add edge messages
// ---------------------------------------------------------------------------
__global__ void sgc_hop_init(const float* __restrict__ zc,
                             const float* __restrict__ dinv,
                             float* __restrict__ zn, int n) {
    int i = blockIdx.x * blockDim.x + threadIdx.x;
    if (i < n) {
        float d = dinv[i];
        zn[i] = d * d * zc[i];
    }
}

__global__ void sgc_hop_edges(const long long* __restrict__ src,
                              const long long* __restrict__ dst,
                              const float* __restrict__ norm,
                              const float* __restrict__ zc,
                              float* __restrict__ zn, int e) {
    int i = blockIdx.x * blockDim.x + threadIdx.x;
    if (i < e) {
        int s = (int)src[i];
        int d = (int)dst[i];
        unsafeAtomicAdd(&zn[d], norm[i] * zc[s]);
    }
}

// 7) out = (z + bias)^2
__global__ void sgc_finalize(const float* __restrict__ z,
                             const float* __restrict__ bias,
                             float* __restrict__ out, int n) {
    int i = blockIdx.x * blockDim.x + threadIdx.x;
    if (i < n) {
        float t = z[i] + bias[0];
        out[i] = t * t;
    }
}

// ---------------------------------------------------------------------------
extern "C" void kernel_launch(void* const* d_in, const int* in_sizes, int n_in,
                              void* d_out, int out_size, void* d_ws, size_t ws_size,
                              hipStream_t stream) {
    const float*     x    = (const float*)d_in[0];
    const long long* ei   = (const long long*)d_in[1];  // int64 [2, E]
    const float*     w    = (const float*)d_in[2];
    const float*     bias = (const float*)d_in[3];
    float*           out  = (float*)d_out;

    const int F = in_sizes[2];           // 128
    const int N = in_sizes[0] / F;       // 50000
    const int E = in_sizes[1] / 2;       // 800000
    const long long* src = ei;           // edge_index[0]
    const long long* dst = ei + E;       // edge_index[1]

    float* dinv = (float*)d_ws;          // N floats (holds deg first, then dinv)
    float* z0   = dinv + N;              // N floats
    float* z1   = z0 + N;                // N floats
    float* norm = z1 + N;                // E floats
    (void)ws_size; (void)n_in; (void)out_size; (void)F;

    const int T = 256;
    const int gN = (N + T - 1) / T;
    const int gE = (E + T - 1) / T;

    // normalization
    sgc_deg_init <<<gN, T, 0, stream>>>(dinv, N);
    sgc_deg_count<<<gE, T, 0, stream>>>(dst, dinv, E);
    sgc_dinv     <<<gN, T, 0, stream>>>(dinv, N);
    sgc_norm     <<<gE, T, 0, stream>>>(src, dst, dinv, norm, E);

    // z = x @ w   (16 rows per wave, 8 waves per block)
    const int tiles  = (N + 15) / 16;
    const int gWmma  = (tiles + 7) / 8;
    sgc_gemv_wmma<<<gWmma, T, 0, stream>>>(x, w, z0, N);

    // K = 3 propagation hops on the projected scalars (ping-pong z0/z1)
    float* zc = z0;
    float* zn = z1;
    for (int k = 0; k < 3; ++k) {
        sgc_hop_init <<<gN, T, 0, stream>>>(zc, dinv, zn, N);
        sgc_hop_edges<<<gE, T, 0, stream>>>(src, dst, norm, zc, zn, E);
        float* t = zc; zc = zn; zn = t;
    }

    // out = (z + b)^2
    sgc_finalize<<<gN, T, 0, stream>>>(zc, bias, out, N);
}